// TripletLossWithMining_11441792877184
// MI455X (gfx1250) — compile-verified
//
#include <hip/hip_runtime.h>
#include <hip/hip_bf16.h>

#define N_PTS 8192
#define DIM   256
#define KSTEP (DIM / 4)        // 64 k-steps of 4 for f32 WMMA
#define MARGIN 0.2f
#define BPAD  260              // LDS row stride in floats: 260 % 64 == 4 -> conflict-free, 16B aligned

typedef float v2f __attribute__((ext_vector_type(2)));
typedef float v8f __attribute__((ext_vector_type(8)));

// ---------------- 1) normalize rows: one wave (32 lanes) per row ----------------
__global__ void normalize_k(const float* __restrict__ emb, float* __restrict__ nrm) {
    int w    = (blockIdx.x * blockDim.x + threadIdx.x) >> 5;
    int lane = threadIdx.x & 31;
    if (w >= N_PTS) return;
    const float* row = emb + (size_t)w * DIM;
    float s = 0.f;
#pragma unroll
    for (int u = 0; u < DIM / 32; ++u) { float v = row[lane + 32 * u]; s += v * v; }
#pragma unroll
    for (int m = 16; m; m >>= 1) s += __shfl_xor(s, m, 32);
    float inv = 1.0f / fmaxf(sqrtf(s), 1e-12f);
    float* o = nrm + (size_t)w * DIM;
#pragma unroll
    for (int u = 0; u < DIM / 32; ++u) o[lane + 32 * u] = row[lane + 32 * u] * inv;
}

// ---- 2) per-anchor: first same-class positive, d_ap, init min arrays ----
__global__ void posap_k(const float* __restrict__ nrm, const int* __restrict__ lab,
                        float* __restrict__ dap_g, float* __restrict__ validf,
                        unsigned* __restrict__ smin_g, unsigned* __restrict__ hmin_g) {
    int i    = (blockIdx.x * blockDim.x + threadIdx.x) >> 5;
    int lane = threadIdx.x & 31;
    if (i >= N_PTS) return;
    int li = lab[i];
    int best = 0x7FFFFFFF;                    // min same-class index != i  == argmax(first true)
    for (int j = lane; j < N_PTS; j += 32)
        if (j != i && lab[j] == li && j < best) best = j;
#pragma unroll
    for (int m = 16; m; m >>= 1) { int b2 = __shfl_xor(best, m, 32); best = best < b2 ? best : b2; }
    int valid = best < N_PTS;
    int pos   = valid ? best : 0;             // matches jnp.argmax(all-false) == 0
    const float* a = nrm + (size_t)i * DIM;
    const float* b = nrm + (size_t)pos * DIM;
    float s = 0.f;
#pragma unroll
    for (int u = 0; u < DIM / 32; ++u) s += a[lane + 32 * u] * b[lane + 32 * u];
#pragma unroll
    for (int m = 16; m; m >>= 1) s += __shfl_xor(s, m, 32);
    if (lane == 0) {
        dap_g[i]  = fminf(fmaxf(1.0f - s, 0.0f), 2.0f);
        validf[i] = valid ? 1.0f : 0.0f;
        smin_g[i] = 0x7F7FFFFFu;              // FLT_MAX bit pattern (dist >= 0 -> u32 order == float order)
        hmin_g[i] = 0x7F7FFFFFu;
    }
}

// ---------------- 3) fused X·X^T + semi-hard mining ----------------
__device__ __forceinline__ void load_b_tile(float* dst, const float* __restrict__ X,
                                            int col0, int tid) {
    // 256 threads stage 16 cols x 256 k of fp32 into LDS (stride BPAD)
    int c  = tid >> 4;                 // 0..15 : which column (row of X)
    int k0 = (tid & 15) * 16;          // 0..240
    const float* src = X + (size_t)(col0 + c) * DIM + k0;
    float* d = dst + c * BPAD + k0;
#pragma unroll
    for (int q = 0; q < 4; ++q)
        *(float4*)(d + 4 * q) = *(const float4*)(src + 4 * q);
}

__global__ __launch_bounds__(256) void mine_k(const float* __restrict__ X,
                                              const int* __restrict__ lab,
                                              const float* __restrict__ dap_g,
                                              unsigned* __restrict__ smin_g,
                                              unsigned* __restrict__ hmin_g) {
    __shared__ float Bsm[2][16 * BPAD];       // double-buffered B tile, 33 KB
    const int tid  = threadIdx.x;
    const int wv   = tid >> 5, lane = tid & 31;
    const int hi   = lane >> 4, ln = lane & 15;
    const int R0   = blockIdx.x * 128 + wv * 16;   // this wave's 16 anchor rows
    const int C0   = blockIdx.y * 1024;            // this block's 1024-column chunk

    // A fragment resident in VGPRs for the whole pass.
    // f32 16x16x4 A layout: lane holds row M=ln, K = 4t + 2*hi + {0,1}.
    v2f afr[KSTEP];
    {
        const float* ar = X + (size_t)(R0 + ln) * DIM + 2 * hi;
#pragma unroll
        for (int t = 0; t < KSTEP; ++t) afr[t] = *(const v2f*)(ar + 4 * t);
    }
    // C layout: VGPR r holds (M = r + 8*hi, N = ln)
    int labR[8]; float dapR[8];
#pragma unroll
    for (int r = 0; r < 8; ++r) { int rr = R0 + 8 * hi + r; labR[r] = lab[rr]; dapR[r] = dap_g[rr]; }

    unsigned sMin[8], hMin[8];
#pragma unroll
    for (int r = 0; r < 8; ++r) { sMin[r] = 0x7F7FFFFFu; hMin[r] = 0x7F7FFFFFu; }

    load_b_tile(Bsm[0], X, C0, tid);
    __syncthreads();

    for (int jt = 0; jt < 64; ++jt) {
        const int cur = jt & 1;
        if (jt + 1 < 64) load_b_tile(Bsm[cur ^ 1], X, C0 + (jt + 1) * 16, tid);

        const int col  = C0 + jt * 16 + ln;
        const int labC = lab[col];
        const float* bb = &Bsm[cur][ln * BPAD + 2 * hi];

        v8f c0 = {}; v8f c1 = {};                 // two accumulators break D->C chain
#pragma unroll
        for (int t = 0; t < KSTEP; t += 2) {
            v2f b0 = *(const v2f*)(bb + 4 * t);
            v2f b1 = *(const v2f*)(bb + 4 * t + 4);
            c0 = __builtin_amdgcn_wmma_f32_16x16x4_f32(false, afr[t],     false, b0, (short)0, c0, false, false);
            c1 = __builtin_amdgcn_wmma_f32_16x16x4_f32(false, afr[t + 1], false, b1, (short)0, c1, false, false);
        }
        v8f cs = c0 + c1;

#pragma unroll
        for (int r = 0; r < 8; ++r) {
            float dist = fminf(fmaxf(1.0f - cs[r], 0.0f), 2.0f);
            if (labR[r] != labC) {                // negatives; diagonal is same-class -> excluded
                unsigned db = __float_as_uint(dist);
                hMin[r] = hMin[r] < db ? hMin[r] : db;
                if (dist > dapR[r] && dist < dapR[r] + MARGIN)
                    sMin[r] = sMin[r] < db ? sMin[r] : db;
            }
        }
        __syncthreads();                          // done reading cur + done writing nxt
    }

    // reduce across the 16 lanes sharing each row set, then one atomic per row
#pragma unroll
    for (int r = 0; r < 8; ++r) {
        unsigned s = sMin[r], h = hMin[r];
#pragma unroll
        for (int m = 8; m; m >>= 1) {
            unsigned s2 = __shfl_xor(s, m, 32);
            unsigned h2 = __shfl_xor(h, m, 32);
            s = s < s2 ? s : s2;
            h = h < h2 ? h : h2;
        }
        if (ln == r) {
            int rr = R0 + 8 * hi + r;
            atomicMin(&smin_g[rr], s);
            atomicMin(&hmin_g[rr], h);
        }
    }
}

// ---------------- 4) final masked mean ----------------
__global__ void loss_k(const float* __restrict__ dap_g, const float* __restrict__ validf,
                       const unsigned* __restrict__ smin_g, const unsigned* __restrict__ hmin_g,
                       float* __restrict__ out) {
    __shared__ float sh_sum[256], sh_cnt[256];
    float sum = 0.f, cnt = 0.f;
    for (int i = threadIdx.x; i < N_PTS; i += 256) {
        float v  = validf[i];
        float ds = __uint_as_float(smin_g[i]);
        float dh = __uint_as_float(hmin_g[i]);
        float dan = (ds < 3.0f) ? ds : dh;        // semi-hard dists < 2.2; sentinel is FLT_MAX
        sum += fmaxf(dap_g[i] - dan + MARGIN, 0.0f) * v;
        cnt += v;
    }
    sh_sum[threadIdx.x] = sum; sh_cnt[threadIdx.x] = cnt;
    __syncthreads();
    for (int s = 128; s; s >>= 1) {
        if (threadIdx.x < s) {
            sh_sum[threadIdx.x] += sh_sum[threadIdx.x + s];
            sh_cnt[threadIdx.x] += sh_cnt[threadIdx.x + s];
        }
        __syncthreads();
    }
    if (threadIdx.x == 0) out[0] = sh_cnt[0] > 0.f ? sh_sum[0] / sh_cnt[0] : 0.f;
}

extern "C" void kernel_launch(void* const* d_in, const int* in_sizes, int n_in,
                              void* d_out, int out_size, void* d_ws, size_t ws_size,
                              hipStream_t stream) {
    const float* emb = (const float*)d_in[0];
    const int*   lab = (const int*)d_in[1];
    float*       out = (float*)d_out;

    char* ws = (char*)d_ws;
    float*    nrm    = (float*)ws;                                    // 8 MB normalized X
    float*    dap    = (float*)(ws + (size_t)N_PTS * DIM * sizeof(float));
    float*    validf = dap + N_PTS;
    unsigned* smin   = (unsigned*)(validf + N_PTS);
    unsigned* hmin   = smin + N_PTS;

    normalize_k<<<(N_PTS * 32) / 256, 256, 0, stream>>>(emb, nrm);
    posap_k<<<(N_PTS * 32) / 256, 256, 0, stream>>>(nrm, lab, dap, validf, smin, hmin);
    dim3 g(N_PTS / 128, N_PTS / 1024);        // 64 row-blocks x 8 column-splits
    mine_k<<<g, 256, 0, stream>>>(nrm, lab, dap, smin, hmin);
    loss_k<<<1, 256, 0, stream>>>(dap, validf, smin, hmin, out);
}